// GCN2_82291573391436
// MI455X (gfx1250) — compile-verified
//
#include <hip/hip_runtime.h>
#include <hip/hip_bf16.h>
#include <float.h>

#define NN 100000      // nodes
#define D  128         // feature dim (= hidden)
#define NG 512         // graphs
#define NC 2           // classes

typedef float v2f __attribute__((ext_vector_type(2)));
typedef float v8f __attribute__((ext_vector_type(8)));

// ============ CSR build: counting sort of edges by dst ============

__global__ void zero_int_kernel(int* __restrict__ p, int n) {
    int i = blockIdx.x * blockDim.x + threadIdx.x;
    if (i < n) p[i] = 0;
}

__global__ void hist_kernel(const int* __restrict__ dst, int* __restrict__ cnt, int e) {
    int i = blockIdx.x * blockDim.x + threadIdx.x;
    if (i < e) atomicAdd(&cnt[dst[i]], 1);
}

// dinv[i] = 1/sqrt(in_degree + 1)   (self-loop included)
__global__ void dinv_kernel(const int* __restrict__ cnt, float* __restrict__ dinv, int n) {
    int i = blockIdx.x * blockDim.x + threadIdx.x;
    if (i < n) dinv[i] = rsqrtf((float)(cnt[i] + 1));
}

__global__ void block_sum_kernel(const int* __restrict__ cnt, int* __restrict__ bsum, int n) {
    __shared__ int sdata[256];
    int i = blockIdx.x * 256 + threadIdx.x;
    sdata[threadIdx.x] = (i < n) ? cnt[i] : 0;
    __syncthreads();
    for (int s = 128; s > 0; s >>= 1) {
        if ((int)threadIdx.x < s) sdata[threadIdx.x] += sdata[threadIdx.x + s];
        __syncthreads();
    }
    if (threadIdx.x == 0) bsum[blockIdx.x] = sdata[0];
}

// serial exclusive scan of ~391 block sums (one thread; trivial work)
__global__ void scan_partials_kernel(int* __restrict__ bsum, int nb,
                                     int* __restrict__ rowptr, int n, int e) {
    if (blockIdx.x == 0 && threadIdx.x == 0) {
        int run = 0;
        for (int i = 0; i < nb; ++i) { int v = bsum[i]; bsum[i] = run; run += v; }
        rowptr[n] = e;
    }
}

// each thread serially scans its 256-chunk, writing exclusive prefix into rowptr
__global__ void chunk_scan_kernel(const int* __restrict__ cnt, const int* __restrict__ bsum,
                                  int* __restrict__ rowptr, int n) {
    int t = blockIdx.x * blockDim.x + threadIdx.x;
    int nb = (n + 255) / 256;
    if (t >= nb) return;
    int run = bsum[t];
    int base = t * 256;
    int lim  = min(base + 256, n);
    for (int i = base; i < lim; ++i) { rowptr[i] = run; run += cnt[i]; }
}

// packed CSR entry: .x = source node, .y = bit pattern of edge weight
__global__ void csr_fill_kernel(const int* __restrict__ src, const int* __restrict__ dst,
                                const float* __restrict__ dinv, const int* __restrict__ rowptr,
                                int* __restrict__ fill, int2* __restrict__ csr, int e) {
    int i = blockIdx.x * blockDim.x + threadIdx.x;
    if (i >= e) return;
    int d = dst[i], s = src[i];
    int pos = rowptr[d] + atomicAdd(&fill[d], 1);
    int2 ent;
    ent.x = s;
    ent.y = __float_as_int(dinv[s] * dinv[d]);
    csr[pos] = ent;
}

// ============ WMMA fp32 GEMM: Y = act(A) @ W,  A:[N,128] W:[128,128] ============
// Block = 256 threads = 8 waves; block owns a 16-row strip; wave w owns cols [16w,16w+16).
// 32 x v_wmma_f32_16x16x4_f32 per wave over K=128. EXEC all-ones (N % 16 == 0, full blocks).

template <bool RELU>
__global__ void __launch_bounds__(256)
gemm_wmma_f32_kernel(const float* __restrict__ A, const float* __restrict__ W,
                     float* __restrict__ Y) {
    const int lane = threadIdx.x & 31;
    const int wave = threadIdx.x >> 5;       // 0..7 -> N tile
    const int half = lane >> 4;              // K split across lane halves
    const int l16  = lane & 15;
    const int row0 = blockIdx.x * 16;
    const int ncol = wave * 16 + l16;
    const size_t abase = (size_t)(row0 + l16) * D;

    v8f acc = {};
    #pragma unroll 4
    for (int k = 0; k < D; k += 4) {
        const int ka = k + 2 * half;
        // A fragment (16x4): lane<16 -> M=l16, K=k,k+1 ; lanes>=16 -> K=k+2,k+3
        v2f av = *(const v2f*)(A + abase + ka);       // 8B aligned (ka even)
        if (RELU) { av.x = fmaxf(av.x, 0.0f); av.y = fmaxf(av.y, 0.0f); }
        // B fragment (4x16): col N = ncol, rows ka, ka+1
        v2f bv; bv.x = W[(size_t)ka * D + ncol];
                bv.y = W[(size_t)(ka + 1) * D + ncol];
        acc = __builtin_amdgcn_wmma_f32_16x16x4_f32(
                  false, av, false, bv, (short)0, acc, false, false);
    }
    const int crow = row0 + half * 8;
    #pragma unroll
    for (int r = 0; r < 8; ++r)
        Y[(size_t)(crow + r) * D + wave * 16 + l16] = acc[r];
}

// ============ pull-based aggregation (no atomics) ============
// X[i,:] = b + Y[i,:]*dinv[i]^2 + sum_{e in in(i)} w[e] * Y[src[e],:]
// Edge loop unrolled x4: 4 independent packed-entry loads then 4 independent
// 512B row gathers in flight per wave for memory-level parallelism.

__global__ void __launch_bounds__(256)
gcn_pull_kernel(const float* __restrict__ Y, const int* __restrict__ rowptr,
                const int2* __restrict__ csr, const float* __restrict__ dinv,
                const float* __restrict__ bias, float* __restrict__ X, int n) {
    int node = blockIdx.x * 8 + (threadIdx.x >> 5);
    if (node >= n) return;
    node = __builtin_amdgcn_readfirstlane(node);     // uniform per wave -> SGPR
    const int lane = threadIdx.x & 31;

    const float s  = dinv[node];
    const float sl = s * s;
    const float4 bv = ((const float4*)bias)[lane];
    float4 acc = ((const float4*)(Y + (size_t)node * D))[lane];
    acc.x = bv.x + acc.x * sl;  acc.y = bv.y + acc.y * sl;
    acc.z = bv.z + acc.z * sl;  acc.w = bv.w + acc.w * sl;

    const int beg = rowptr[node];
    const int end = rowptr[node + 1];
    int e = beg;
    for (; e + 4 <= end; e += 4) {
        const int2 p0 = csr[e + 0];
        const int2 p1 = csr[e + 1];
        const int2 p2 = csr[e + 2];
        const int2 p3 = csr[e + 3];
        const float4 v0 = ((const float4*)(Y + (size_t)p0.x * D))[lane];
        const float4 v1 = ((const float4*)(Y + (size_t)p1.x * D))[lane];
        const float4 v2 = ((const float4*)(Y + (size_t)p2.x * D))[lane];
        const float4 v3 = ((const float4*)(Y + (size_t)p3.x * D))[lane];
        const float w0 = __int_as_float(p0.y), w1 = __int_as_float(p1.y);
        const float w2 = __int_as_float(p2.y), w3 = __int_as_float(p3.y);
        acc.x += v0.x * w0;  acc.y += v0.y * w0;  acc.z += v0.z * w0;  acc.w += v0.w * w0;
        acc.x += v1.x * w1;  acc.y += v1.y * w1;  acc.z += v1.z * w1;  acc.w += v1.w * w1;
        acc.x += v2.x * w2;  acc.y += v2.y * w2;  acc.z += v2.z * w2;  acc.w += v2.w * w2;
        acc.x += v3.x * w3;  acc.y += v3.y * w3;  acc.z += v3.z * w3;  acc.w += v3.w * w3;
    }
    for (; e < end; ++e) {
        const int2 p = csr[e];
        const float w = __int_as_float(p.y);
        const float4 v = ((const float4*)(Y + (size_t)p.x * D))[lane];
        acc.x += v.x * w;  acc.y += v.y * w;  acc.z += v.z * w;  acc.w += v.w * w;
    }
    ((float4*)(X + (size_t)node * D))[lane] = acc;
}

// ============ pooling via sorted-batch segment boundaries (no atomics) ============

__global__ void graph_bounds_kernel(const int* __restrict__ batch, int* __restrict__ starts,
                                    int n) {
    int g = blockIdx.x * blockDim.x + threadIdx.x;
    if (g > NG) return;
    if (g == NG) { starts[NG] = n; return; }
    int lo = 0, hi = n;                              // lower_bound(batch, g)
    while (lo < hi) { int mid = (lo + hi) >> 1; if (batch[mid] < g) lo = mid + 1; else hi = mid; }
    starts[g] = lo;
}

__global__ void __launch_bounds__(256)
pool_kernel(const float* __restrict__ H, const int* __restrict__ starts,
            float* __restrict__ sums, float* __restrict__ maxs, float* __restrict__ cnts) {
    int g = blockIdx.x * 8 + (threadIdx.x >> 5);
    if (g >= NG) return;
    g = __builtin_amdgcn_readfirstlane(g);
    const int lane = threadIdx.x & 31;
    const int beg = starts[g], end = starts[g + 1];
    float4 s = {0.0f, 0.0f, 0.0f, 0.0f};
    float4 m = {-FLT_MAX, -FLT_MAX, -FLT_MAX, -FLT_MAX};
    int i = beg;
    for (; i + 2 <= end; i += 2) {
        const float4 v0 = ((const float4*)(H + (size_t)(i + 0) * D))[lane];
        const float4 v1 = ((const float4*)(H + (size_t)(i + 1) * D))[lane];
        s.x += v0.x + v1.x;  s.y += v0.y + v1.y;
        s.z += v0.z + v1.z;  s.w += v0.w + v1.w;
        m.x = fmaxf(m.x, fmaxf(v0.x, v1.x));  m.y = fmaxf(m.y, fmaxf(v0.y, v1.y));
        m.z = fmaxf(m.z, fmaxf(v0.z, v1.z));  m.w = fmaxf(m.w, fmaxf(v0.w, v1.w));
    }
    for (; i < end; ++i) {
        const float4 v = ((const float4*)(H + (size_t)i * D))[lane];
        s.x += v.x; s.y += v.y; s.z += v.z; s.w += v.w;
        m.x = fmaxf(m.x, v.x); m.y = fmaxf(m.y, v.y);
        m.z = fmaxf(m.z, v.z); m.w = fmaxf(m.w, v.w);
    }
    ((float4*)(sums + (size_t)g * D))[lane] = s;
    ((float4*)(maxs + (size_t)g * D))[lane] = m;
    if (lane == 0) cnts[g] = (float)(end - beg);
}

// ============ classifier: out[g,c] = [mean || max] @ Wc + bc ============

__global__ void classifier_kernel(const float* __restrict__ sums, const float* __restrict__ maxs,
                                  const float* __restrict__ cnts, const float* __restrict__ Wc,
                                  const float* __restrict__ bc, float* __restrict__ out) {
    int t = blockIdx.x * blockDim.x + threadIdx.x;
    if (t >= NG * NC) return;
    int g = t / NC, c = t % NC;
    float inv = 1.0f / fmaxf(cnts[g], 1.0f);
    float acc = bc[c];
    #pragma unroll 4
    for (int k = 0; k < D; ++k)
        acc += (sums[(size_t)g * D + k] * inv) * Wc[k * NC + c];
    #pragma unroll 4
    for (int k = 0; k < D; ++k)
        acc += maxs[(size_t)g * D + k] * Wc[(D + k) * NC + c];
    out[t] = acc;
}

// ============ launch ============

static inline size_t align4(size_t x) { return (x + 3) & ~(size_t)3; }  // 16B in elements

extern "C" void kernel_launch(void* const* d_in, const int* in_sizes, int n_in,
                              void* d_out, int out_size, void* d_ws, size_t ws_size,
                              hipStream_t stream) {
    const float* x     = (const float*)d_in[0];
    const int*   ei    = (const int*)d_in[1];
    const int*   batch = (const int*)d_in[2];
    const float* W1    = (const float*)d_in[3];
    const float* b1    = (const float*)d_in[4];
    const float* W2    = (const float*)d_in[5];
    const float* b2    = (const float*)d_in[6];
    const float* W3    = (const float*)d_in[7];
    const float* b3    = (const float*)d_in[8];
    const float* Wc    = (const float*)d_in[9];
    const float* bc    = (const float*)d_in[10];
    float* out = (float*)d_out;

    const int E = in_sizes[1] / 2;
    const int* src = ei;
    const int* dst = ei + E;
    const int nb = (NN + 255) / 256;                // #scan chunks

    // ---- workspace carve-up (all offsets kept 16B aligned) ----
    char* base = (char*)d_ws;
    size_t off = 0;
    float* dinv    = (float*)(base + off); off += align4(NN)       * 4;
    int*   fill    = (int*)  (base + off); off += align4(NN)       * 4;  // hist / fill counters
    int*   rowptr  = (int*)  (base + off); off += align4(NN + 1)   * 4;
    int*   bsum    = (int*)  (base + off); off += align4(nb)       * 4;
    int*   starts  = (int*)  (base + off); off += align4(NG + 1)   * 4;
    int2*  csr     = (int2*) (base + off); off += (size_t)E * 2    * 4;  // packed {src, w}
    float* hA      = (float*)(base + off); off += (size_t)NN * D   * 4;
    float* hB      = (float*)(base + off); off += (size_t)NN * D   * 4;
    float* sums    = (float*)(base + off); off += (size_t)NG * D   * 4;
    float* maxs    = (float*)(base + off); off += (size_t)NG * D   * 4;
    float* cnts    = (float*)(base + off); off += align4(NG)       * 4;

    const int T  = 256;
    const int gN = (NN + T - 1) / T;
    const int gE = (E + T - 1) / T;

    // ---- CSR build (counting sort by dst) + dinv ----
    zero_int_kernel<<<gN, T, 0, stream>>>(fill, NN);
    hist_kernel<<<gE, T, 0, stream>>>(dst, fill, E);
    dinv_kernel<<<gN, T, 0, stream>>>(fill, dinv, NN);
    block_sum_kernel<<<nb, T, 0, stream>>>(fill, bsum, NN);
    scan_partials_kernel<<<1, 1, 0, stream>>>(bsum, nb, rowptr, NN, E);
    chunk_scan_kernel<<<(nb + T - 1) / T, T, 0, stream>>>(fill, bsum, rowptr, NN);
    zero_int_kernel<<<gN, T, 0, stream>>>(fill, NN);
    csr_fill_kernel<<<gE, T, 0, stream>>>(src, dst, dinv, rowptr, fill, csr, E);

    // ---- 3 GCN layers: WMMA GEMM + pull aggregation ----
    const int gMM = NN / 16;           // 6250 M-tile blocks
    const int gNw = (NN + 7) / 8;      // wave per node

    gemm_wmma_f32_kernel<false><<<gMM, T, 0, stream>>>(x, W1, hB);
    gcn_pull_kernel<<<gNw, T, 0, stream>>>(hB, rowptr, csr, dinv, b1, hA, NN);

    gemm_wmma_f32_kernel<true><<<gMM, T, 0, stream>>>(hA, W2, hB);
    gcn_pull_kernel<<<gNw, T, 0, stream>>>(hB, rowptr, csr, dinv, b2, hA, NN);

    gemm_wmma_f32_kernel<true><<<gMM, T, 0, stream>>>(hA, W3, hB);
    gcn_pull_kernel<<<gNw, T, 0, stream>>>(hB, rowptr, csr, dinv, b3, hA, NN);

    // ---- pooling + classifier ----
    graph_bounds_kernel<<<(NG + 1 + T - 1) / T, T, 0, stream>>>(batch, starts, NN);
    pool_kernel<<<(NG + 7) / 8, T, 0, stream>>>(hA, starts, sums, maxs, cnts);
    classifier_kernel<<<(NG * NC + T - 1) / T, T, 0, stream>>>(sums, maxs, cnts, Wc, bc, out);
}